// Model_16827681865991
// MI455X (gfx1250) — compile-verified
//
#include <hip/hip_runtime.h>

#define SIZE 768
#define T_STEPS 24
#define BATCH 32

typedef __attribute__((ext_vector_type(2))) float v2f;
typedef __attribute__((ext_vector_type(8))) float v8f;

// y0[b,i] = param[i] + 1.0
__global__ void init_y_kernel(const float* __restrict__ param, float* __restrict__ y0) {
    int idx = blockIdx.x * blockDim.x + threadIdx.x;
    if (idx < BATCH * SIZE) {
        y0[idx] = param[idx % SIZE] + 1.0f;
    }
}

// One step of the scan: y_pre[b,k] = sum_i y_in[b,i] * inp[b,t,i,k]
// y_carry[b,k] = relu(y_pre[b,k]); y_pre written to `out` only on last step.
// Each wave computes a 16-column tile via V_WMMA_F32_16X16X4_F32 with the
// y-chunk replicated across all 16 A-rows (M=1 matvec padded to M=16).
__global__ __launch_bounds__(256) void step_kernel(
    const float* __restrict__ inp, const float* __restrict__ y_in,
    float* __restrict__ y_carry, float* __restrict__ y_pre,
    int t, int write_pre) {
    __shared__ float ly[SIZE];

    const int b    = blockIdx.y;
    const int tid  = threadIdx.x;   // 0..255
    const int wave = tid >> 5;      // 0..7
    const int lane = tid & 31;
    const int n    = lane & 15;     // column within 16-wide tile
    const int hi   = lane >> 4;     // half-wave selects K pairs {0,1} vs {2,3}

    // Stage y[b, :] (3 KB) into LDS once; broadcast-read in the K loop.
    for (int i = tid; i < SIZE; i += 256) ly[i] = y_in[b * SIZE + i];
    __syncthreads();

    const float* M = inp + ((size_t)b * T_STEPS + t) * (size_t)SIZE * SIZE;
    const int col = blockIdx.x * 128 + wave * 16 + n;
    const float* colp = M + col;

    v8f c = {};  // f32 16x16 accumulator (8 VGPRs)

#pragma unroll 8
    for (int i0 = 0; i0 < SIZE; i0 += 4) {
        const int ib = i0 + 2 * hi;
        // A fragment: 16x4 f32, all rows = y chunk. VGPR0 = K(0|2), VGPR1 = K(1|3).
        v2f A = *(const v2f*)&ly[ib];
        // B fragment: 4x16 f32, row K striped across lanes; mirrored half-wave split.
        v2f Bf;
        Bf.x = colp[(size_t)ib * SIZE];        // M[ib,   col]
        Bf.y = colp[(size_t)(ib + 1) * SIZE];  // M[ib+1, col]
        // 8 args: (neg_a, A, neg_b, B, c_mod, C, reuse_a, reuse_b)
        c = __builtin_amdgcn_wmma_f32_16x16x4_f32(
            false, A, false, Bf, (short)0, c, false, false);
    }

    // All D rows are identical (replicated A); row M=0 lives in c[0] on lanes 0-15.
    if (hi == 0) {
        float y = c[0];
        y_carry[b * SIZE + col] = y > 0.0f ? y : 0.0f;
        if (write_pre) y_pre[b * SIZE + col] = y;
    }
}

extern "C" void kernel_launch(void* const* d_in, const int* in_sizes, int n_in,
                              void* d_out, int out_size, void* d_ws, size_t ws_size,
                              hipStream_t stream) {
    const float* inp   = (const float*)d_in[0];  // [B, T, SIZE, SIZE] f32
    const float* param = (const float*)d_in[1];  // [SIZE] f32
    float* out = (float*)d_out;                  // [B, SIZE] f32 (pre-ReLU last step)

    // Ping-pong carry buffers in workspace: 2 * 32 * 768 * 4 = 192 KB.
    float* y0 = (float*)d_ws;
    float* y1 = y0 + BATCH * SIZE;

    init_y_kernel<<<(BATCH * SIZE + 255) / 256, 256, 0, stream>>>(param, y0);

    dim3 grid(SIZE / 128, BATCH);  // 6 x 32 blocks, 8 waves each
    for (int t = 0; t < T_STEPS; ++t) {
        const float* yin  = (t & 1) ? y1 : y0;
        float*       ycar = (t & 1) ? y0 : y1;
        const int    last = (t == T_STEPS - 1);
        step_kernel<<<grid, 256, 0, stream>>>(inp, yin, ycar, out, t, last);
    }
}